// Net_45131516346671
// MI455X (gfx1250) — compile-verified
//
#include <hip/hip_runtime.h>
#include <hip/hip_bf16.h>

typedef float v2f __attribute__((ext_vector_type(2)));
typedef float v8f __attribute__((ext_vector_type(8)));

constexpr long long NN = 100000;   // nodes
constexpr long long EE = 1600000;  // edges (without self loops)
constexpr long long ET = 1700000;  // edges + self loops
constexpr int FD = 128;            // feature dim (HID == F_IN == 128)
constexpr int NG = 512;            // graphs
constexpr int NH = 64;             // fc hidden
constexpr int NC = 10;             // classes

__device__ __forceinline__ float selu_f(float x) {
  const float sc = 1.0507009873554805f;
  const float al = 1.6732632423543772f;
  return x > 0.f ? sc * x : sc * al * (__expf(x) - 1.f);
}

// ---------------------------------------------------------------------------
// GEMM: H[n x 128] = X[n x 128] @ W[128 x 128], fp32 WMMA 16x16x4.
// One block = 16 output rows, 8 waves = 8 column tiles of 16.
// ---------------------------------------------------------------------------
__global__ __launch_bounds__(256) void gat_gemm128(const float* __restrict__ X,
                                                   const float* __restrict__ W,
                                                   float* __restrict__ H, int n) {
  __shared__ float As[16][132];                    // padded: stride 132 kills bank conflicts
  __shared__ __align__(16) float Ws[128][128];     // full weight matrix, 64 KB
  const int t = threadIdx.x;
  const int rowBase = blockIdx.x * 16;

  // Stage X tile: 2048 floats, 8 per thread
  for (int i = 0; i < 8; ++i) {
    int idx = t * 8 + i;
    int r = idx >> 7, c = idx & 127;
    int row = rowBase + r;
    As[r][c] = (row < n) ? X[(long long)row * FD + c] : 0.0f;
  }
  // Stage W: 16384 floats as float4, 16 per thread
  const float4* W4 = (const float4*)W;
  float4* Ws4 = (float4*)&Ws[0][0];
  for (int i = 0; i < 16; ++i) {
    int idx = t + i * 256;
    Ws4[idx] = W4[idx];
  }
  __syncthreads();

  const int lane = t & 31;
  const int wave = t >> 5;
  const int colBase = wave * 16;
  const int m  = lane & 15;          // M (A/D row sub-index) / N for B,C
  const int kh = (lane >> 4) * 2;    // K pair selector (lanes 16-31 hold K+2,K+3)

  v8f acc = {};
#pragma unroll
  for (int k0 = 0; k0 < 128; k0 += 4) {
    v2f a, b;
    a.x = As[m][k0 + kh];
    a.y = As[m][k0 + kh + 1];
    b.x = Ws[k0 + kh][colBase + m];
    b.y = Ws[k0 + kh + 1][colBase + m];
    acc = __builtin_amdgcn_wmma_f32_16x16x4_f32(false, a, false, b, (short)0,
                                                acc, false, false);
  }
#pragma unroll
  for (int r = 0; r < 8; ++r) {
    int row = rowBase + r + (lane >> 4) * 8;
    if (row < n) H[(long long)row * FD + colBase + m] = acc[r];
  }
}

// ---------------------------------------------------------------------------
// Per-node attention logits: alpha_s[i] = h[i].a_src ; alpha_d[i] = h[i].a_dst
// One wave32 per node; float4 per lane; shfl reduction.
// ---------------------------------------------------------------------------
__global__ __launch_bounds__(256) void gat_alpha(const float* __restrict__ H,
                                                 const float* __restrict__ aS,
                                                 const float* __restrict__ aD,
                                                 float* __restrict__ oS,
                                                 float* __restrict__ oD, int n) {
  int wid = blockIdx.x * 8 + (threadIdx.x >> 5);
  if (wid >= n) return;
  int lane = threadIdx.x & 31;
  const float4 hv = ((const float4*)(H + (long long)wid * FD))[lane];
  const float4 sv = ((const float4*)aS)[lane];
  const float4 dv = ((const float4*)aD)[lane];
  float vs = hv.x * sv.x + hv.y * sv.y + hv.z * sv.z + hv.w * sv.w;
  float vd = hv.x * dv.x + hv.y * dv.y + hv.z * dv.z + hv.w * dv.w;
  for (int o = 16; o > 0; o >>= 1) {
    vs += __shfl_down(vs, o, 32);
    vd += __shfl_down(vd, o, 32);
  }
  if (lane == 0) { oS[wid] = vs; oD[wid] = vd; }
}

__global__ void gat_fill_u32(unsigned* __restrict__ p, unsigned v, long long n) {
  long long i = (long long)blockIdx.x * 256 + threadIdx.x;
  if (i < n) p[i] = v;
}

__device__ __forceinline__ void edge_sd(const int* __restrict__ ei, long long e,
                                        int& s, int& d) {
  if (e < EE) { s = ei[e]; d = ei[EE + e]; }
  else        { s = d = (int)(e - EE); }
}

// Pass 1: per-destination max of LeakyReLU(alpha_s[src]+alpha_d[dst]),
// via order-preserving uint atomicMax.
__global__ __launch_bounds__(256) void gat_edge_max(const int* __restrict__ ei,
                                                    const float* __restrict__ aS,
                                                    const float* __restrict__ aD,
                                                    unsigned* __restrict__ mk) {
  long long e = (long long)blockIdx.x * 256 + threadIdx.x;
  if (e >= ET) return;
  int s, d; edge_sd(ei, e, s, d);
  float v = aS[s] + aD[d];
  v = v > 0.f ? v : 0.2f * v;
  unsigned u = __float_as_uint(v);
  unsigned key = (u & 0x80000000u) ? ~u : (u | 0x80000000u);
  atomicMax(&mk[d], key);
}

// Pass 2: ex = exp(e - m[dst]); denom[dst] += ex; cache ex.
__global__ __launch_bounds__(256) void gat_edge_exp(const int* __restrict__ ei,
                                                    const float* __restrict__ aS,
                                                    const float* __restrict__ aD,
                                                    const unsigned* __restrict__ mk,
                                                    float* __restrict__ denom,
                                                    float* __restrict__ exbuf) {
  long long e = (long long)blockIdx.x * 256 + threadIdx.x;
  if (e >= ET) return;
  int s, d; edge_sd(ei, e, s, d);
  float v = aS[s] + aD[d];
  v = v > 0.f ? v : 0.2f * v;
  unsigned k = mk[d];
  unsigned u = (k & 0x80000000u) ? (k & 0x7fffffffu) : ~k;
  float ex = __expf(v - __uint_as_float(u));
  exbuf[e] = ex;
  atomicAdd(&denom[d], ex);
}

// Pass 3: out[dst] += (ex/denom[dst]) * h[src]. Wave per edge, grid-strided,
// float4/lane, non-returning f32 atomics; prefetch next src row into cache.
__global__ __launch_bounds__(256) void gat_edge_agg(const int* __restrict__ ei,
                                                    const float* __restrict__ H,
                                                    const float* __restrict__ exbuf,
                                                    const float* __restrict__ denom,
                                                    float* __restrict__ out) {
  const int lane = threadIdx.x & 31;
  const long long nwaves = (long long)gridDim.x * 8;
  for (long long e = (long long)blockIdx.x * 8 + (threadIdx.x >> 5); e < ET;
       e += nwaves) {
    int s, d; edge_sd(ei, e, s, d);
    long long en = e + nwaves;
    if (en < ET) {  // prefetch next iteration's gather row (global_prefetch_b8)
      int ns = (en < EE) ? ei[en] : (int)(en - EE);
      __builtin_prefetch(H + (long long)ns * FD + lane * 4, 0, 3);
    }
    float coef = exbuf[e] / denom[d];
    float4 v = ((const float4*)(H + (long long)s * FD))[lane];
    float* o = out + (long long)d * FD + lane * 4;
    atomicAdd(o + 0, coef * v.x);
    atomicAdd(o + 1, coef * v.y);
    atomicAdd(o + 2, coef * v.z);
    atomicAdd(o + 3, coef * v.w);
  }
}

// out = selu(agg + bias), in place. n = NN*FD elements.
__global__ __launch_bounds__(256) void gat_finalize(float* __restrict__ buf,
                                                    const float* __restrict__ b,
                                                    long long n) {
  long long i = (long long)blockIdx.x * 256 + threadIdx.x;
  if (i < n) buf[i] = selu_f(buf[i] + b[i & (FD - 1)]);
}

// Global mean pool scatter: wave per node.
__global__ __launch_bounds__(256) void gat_pool_scatter(const float* __restrict__ H,
                                                        const int* __restrict__ batch,
                                                        float* __restrict__ sums,
                                                        float* __restrict__ cnts,
                                                        int n) {
  int wid = blockIdx.x * 8 + (threadIdx.x >> 5);
  if (wid >= n) return;
  int lane = threadIdx.x & 31;
  int g = batch[wid];
  float4 v = ((const float4*)(H + (long long)wid * FD))[lane];
  float* s = sums + (long long)g * FD + lane * 4;
  atomicAdd(s + 0, v.x);
  atomicAdd(s + 1, v.y);
  atomicAdd(s + 2, v.z);
  atomicAdd(s + 3, v.w);
  if (lane == 0) atomicAdd(&cnts[g], 1.0f);
}

__global__ __launch_bounds__(256) void gat_pooled(const float* __restrict__ sums,
                                                  const float* __restrict__ cnts,
                                                  float* __restrict__ pooled) {
  int i = blockIdx.x * 256 + threadIdx.x;
  if (i < NG * FD) {
    float c = fmaxf(cnts[i >> 7], 1.0f);
    pooled[i] = selu_f(sums[i] / c);
  }
}

// MLP head + log_softmax: one block (64 threads) per graph.
__global__ __launch_bounds__(64) void gat_fc_head(const float* __restrict__ pooled,
                                                  const float* __restrict__ W1,
                                                  const float* __restrict__ b1,
                                                  const float* __restrict__ W2,
                                                  const float* __restrict__ b2,
                                                  float* __restrict__ out) {
  __shared__ float g1[NH];
  __shared__ float lg[NC];
  __shared__ float red[2];
  const int g = blockIdx.x, j = threadIdx.x;
  const float* p = pooled + (long long)g * FD;
  float acc = b1[j];
  for (int k = 0; k < FD; ++k) acc += p[k] * W1[k * NH + j];
  g1[j] = selu_f(acc);
  __syncthreads();
  if (j < NC) {
    float l = b2[j];
    for (int k = 0; k < NH; ++k) l += g1[k] * W2[k * NC + j];
    lg[j] = l;
  }
  __syncthreads();
  if (j == 0) {
    float mx = lg[0];
    for (int k = 1; k < NC; ++k) mx = fmaxf(mx, lg[k]);
    float s = 0.f;
    for (int k = 0; k < NC; ++k) s += __expf(lg[k] - mx);
    red[0] = mx;
    red[1] = __logf(s);
  }
  __syncthreads();
  if (j < NC) out[g * NC + j] = lg[j] - red[0] - red[1];
}

// ---------------------------------------------------------------------------
static inline int blk256(long long n) { return (int)((n + 255) / 256); }

static void gat_layer(const float* in, const float* W, const float* aS,
                      const float* aD, const float* b, const int* ei,
                      float* hbuf, float* aggbuf, float* alpS, float* alpD,
                      unsigned* mkeys, float* denom, float* exbuf,
                      hipStream_t st) {
  gat_gemm128<<<(int)((NN + 15) / 16), 256, 0, st>>>(in, W, hbuf, (int)NN);
  gat_alpha<<<(int)((NN + 7) / 8), 256, 0, st>>>(hbuf, aS, aD, alpS, alpD, (int)NN);
  gat_fill_u32<<<blk256(NN * FD), 256, 0, st>>>((unsigned*)aggbuf, 0u, NN * FD);
  gat_fill_u32<<<blk256(NN), 256, 0, st>>>(mkeys, 0x007FFFFFu, NN);  // key(-inf)
  gat_fill_u32<<<blk256(NN), 256, 0, st>>>((unsigned*)denom, 0u, NN);
  gat_edge_max<<<blk256(ET), 256, 0, st>>>(ei, alpS, alpD, mkeys);
  gat_edge_exp<<<blk256(ET), 256, 0, st>>>(ei, alpS, alpD, mkeys, denom, exbuf);
  gat_edge_agg<<<8192, 256, 0, st>>>(ei, hbuf, exbuf, denom, aggbuf);
  gat_finalize<<<blk256(NN * FD), 256, 0, st>>>(aggbuf, b, NN * FD);
}

extern "C" void kernel_launch(void* const* d_in, const int* in_sizes, int n_in,
                              void* d_out, int out_size, void* d_ws,
                              size_t ws_size, hipStream_t stream) {
  (void)in_sizes; (void)n_in; (void)out_size; (void)ws_size;

  const float* x      = (const float*)d_in[0];
  const int*   ei     = (const int*)d_in[1];
  const int*   batch  = (const int*)d_in[2];
  const float* W1     = (const float*)d_in[3];
  const float* aS1    = (const float*)d_in[4];
  const float* aD1    = (const float*)d_in[5];
  const float* b1     = (const float*)d_in[6];
  const float* W2     = (const float*)d_in[7];
  const float* aS2    = (const float*)d_in[8];
  const float* aD2    = (const float*)d_in[9];
  const float* b2     = (const float*)d_in[10];
  const float* Wfc1   = (const float*)d_in[11];
  const float* bfc1   = (const float*)d_in[12];
  const float* Wfc2   = (const float*)d_in[13];
  const float* bfc2   = (const float*)d_in[14];

  // Carve workspace (256 B aligned slabs)
  char* w = (char*)d_ws;
  size_t off = 0;
  auto carve = [&](size_t bytes) {
    char* p = w + off;
    off += (bytes + 255) & ~(size_t)255;
    return (void*)p;
  };
  float*    bufA   = (float*)carve(NN * FD * 4);   // h (GEMM output)
  float*    bufB   = (float*)carve(NN * FD * 4);   // agg / layer output
  float*    alpS   = (float*)carve(NN * 4);
  float*    alpD   = (float*)carve(NN * 4);
  unsigned* mkeys  = (unsigned*)carve(NN * 4);
  float*    denom  = (float*)carve(NN * 4);
  float*    exbuf  = (float*)carve(ET * 4);
  float*    sums   = (float*)carve((size_t)NG * FD * 4);
  float*    cnts   = (float*)carve(NG * 4);
  float*    pooled = (float*)carve((size_t)NG * FD * 4);

  // Layer 1: x -> bufB ; Layer 2: bufB -> bufB (h staged in bufA each time)
  gat_layer(x,    W1, aS1, aD1, b1, ei, bufA, bufB, alpS, alpD, mkeys, denom,
            exbuf, stream);
  gat_layer(bufB, W2, aS2, aD2, b2, ei, bufA, bufB, alpS, alpD, mkeys, denom,
            exbuf, stream);
  // NOTE: layer-2 aggregates into bufB after its GEMM consumed bufB; stream
  // ordering makes the in-place reuse safe (fill_u32 zero runs after the GEMM).
  // But the gemm for layer 2 reads bufB while writing bufA, and only after the
  // alpha kernel completes do we zero bufB — all serialized on `stream`.

  // Global mean pool + head
  gat_fill_u32<<<blk256((long long)NG * FD), 256, 0, stream>>>((unsigned*)sums, 0u,
                                                               (long long)NG * FD);
  gat_fill_u32<<<blk256(NG), 256, 0, stream>>>((unsigned*)cnts, 0u, NG);
  gat_pool_scatter<<<(int)((NN + 7) / 8), 256, 0, stream>>>(bufB, batch, sums,
                                                            cnts, (int)NN);
  gat_pooled<<<blk256((long long)NG * FD), 256, 0, stream>>>(sums, cnts, pooled);
  gat_fc_head<<<NG, 64, 0, stream>>>(pooled, Wfc1, bfc1, Wfc2, bfc2,
                                     (float*)d_out);
}